// KernelConv_21406117004159
// MI455X (gfx1250) — compile-verified
//
#include <hip/hip_runtime.h>
#include <hip/hip_bf16.h>

#define HWDIM 128
#define TW    64                    // tile width in pixels (4 per thread)
#define TH    16                    // tile height in pixels
#define HALO  2
#define LTW   (TW + 2 * HALO)       // 68
#define LTH   (TH + 2 * HALO)       // 20
#define KS    5
#define RR    4
#define CC    3
#define LDS_ELEMS (CC * LTH * LTW)  // 4080 floats (16320 B)

typedef __attribute__((address_space(1))) int GI;   // global int
typedef __attribute__((address_space(3))) int LI;   // LDS int
typedef float v4f __attribute__((ext_vector_type(4)));

__device__ __forceinline__ void async_g2l_b32(const float* g, float* l) {
#if __has_builtin(__builtin_amdgcn_global_load_async_to_lds_b32)
    __builtin_amdgcn_global_load_async_to_lds_b32(
        (GI*)(void*)const_cast<float*>(g), (LI*)(void*)l, /*offset=*/0, /*cpol=*/0);
#else
    unsigned lds_addr = (unsigned)(__SIZE_TYPE__)(LI*)(void*)l;
    unsigned long long gaddr = (unsigned long long)g;
    asm volatile("global_load_async_to_lds_b32 %0, %1, off"
                 :: "v"(lds_addr), "v"(gaddr) : "memory");
#endif
}

__device__ __forceinline__ void wait_asynccnt0() {
#if __has_builtin(__builtin_amdgcn_s_wait_asynccnt)
    __builtin_amdgcn_s_wait_asynccnt(0);
#else
    asm volatile("s_wait_asynccnt 0" ::: "memory");
#endif
}

__global__ __launch_bounds__(256)
void adaptive_conv_ps_kernel(const float* __restrict__ burst,
                             const float* __restrict__ kernels,
                             float* __restrict__ out) {
    __shared__ float sB[LDS_ELEMS];

    const int tid   = threadIdx.x;
    const int tx    = tid & 15;             // 16 threads across -> 64 pixels (4/thread)
    const int ty    = tid >> 4;             // 16 rows
    const int baseX = blockIdx.x * TW;
    const int baseY = blockIdx.y * TH;
    const int slice = blockIdx.z;           // n*8 + d

    const float* Bs = burst + (size_t)slice * CC * HWDIM * HWDIM;

    // ---- Stage burst tile (zero-padded halo) into LDS via async copies ----
    for (int idx = tid; idx < LDS_ELEMS; idx += 256) {
        int c   = idx / (LTH * LTW);
        int rem = idx - c * (LTH * LTW);
        int row = rem / LTW;
        int col = rem - row * LTW;
        int gy  = baseY + row - HALO;
        int gx  = baseX + col - HALO;
        if ((unsigned)gy < HWDIM && (unsigned)gx < HWDIM) {
            async_g2l_b32(Bs + ((size_t)c * HWDIM + gy) * HWDIM + gx, &sB[idx]);
        } else {
            sB[idx] = 0.0f;                 // zero-pad halo
        }
    }
    wait_asynccnt0();
    __syncthreads();

    // ---- Four pixels per thread: (h, w0 .. w0+3) ----
    const int h  = baseY + ty;
    const int w0 = baseX + 4 * tx;          // 16B-aligned global column
    const int hw = h * HWDIM + w0;
    const int kb = slice * (RR * KS * KS);  // kernels (slice, r, i, j, h, w)

    float acc[CC][4][RR];                   // [channel][pixel][r]
#pragma unroll
    for (int c = 0; c < CC; ++c)
#pragma unroll
        for (int p = 0; p < 4; ++p)
#pragma unroll
            for (int r = 0; r < RR; ++r) acc[c][p][r] = 0.0f;

#pragma unroll
    for (int i = 0; i < KS; ++i) {
        // 8 consecutive patch values per channel cover taps j=0..4 for 4 pixels
        // (16B-aligned -> two ds b128 reads per channel)
        float rowv[CC][8];
#pragma unroll
        for (int c = 0; c < CC; ++c)
#pragma unroll
            for (int q = 0; q < 8; ++q)
                rowv[c][q] = sB[c * (LTH * LTW) + (ty + i) * LTW + 4 * tx + q];

#pragma unroll
        for (int j = 0; j < KS; ++j) {
            const int t = i * KS + j;
            v4f kw[RR];                     // per-pixel weights: 16B coalesced NT loads
#pragma unroll
            for (int r = 0; r < RR; ++r)
                kw[r] = __builtin_nontemporal_load((const v4f*)(
                    kernels + (size_t)(kb + r * KS * KS + t) * (HWDIM * HWDIM) + hw));

#pragma unroll
            for (int r = 0; r < RR; ++r) {
#pragma unroll
                for (int c = 0; c < CC; ++c) {
#pragma unroll
                    for (int p = 0; p < 4; ++p)
                        acc[c][p][r] = fmaf(rowv[c][j + p], kw[r][p], acc[c][p][r]);
                }
            }
        }
    }

    // ---- Pixel-shuffle store: out (slice, c, 2H, 2W); r = si*2 + sj ----
    // Pixels w0..w0+3 produce 8 consecutive output columns: two 16B NT stores each row.
    float* Os = out + (size_t)slice * CC * (2 * HWDIM) * (2 * HWDIM);
#pragma unroll
    for (int c = 0; c < CC; ++c) {
#pragma unroll
        for (int si = 0; si < 2; ++si) {
            float* rowp = Os + ((size_t)c * (2 * HWDIM) + (2 * h + si)) * (2 * HWDIM)
                          + 2 * w0;
#pragma unroll
            for (int half = 0; half < 2; ++half) {
                v4f v;
                v.x = acc[c][2 * half + 0][si * 2 + 0];
                v.y = acc[c][2 * half + 0][si * 2 + 1];
                v.z = acc[c][2 * half + 1][si * 2 + 0];
                v.w = acc[c][2 * half + 1][si * 2 + 1];
                __builtin_nontemporal_store(v, (v4f*)(rowp + 4 * half));
            }
        }
    }
}

extern "C" void kernel_launch(void* const* d_in, const int* in_sizes, int n_in,
                              void* d_out, int out_size, void* d_ws, size_t ws_size,
                              hipStream_t stream) {
    (void)n_in; (void)out_size; (void)d_ws; (void)ws_size;
    const float* burst   = (const float*)d_in[0];
    const float* kernels = (const float*)d_in[1];
    float* out = (float*)d_out;

    const int slices = in_sizes[0] / (CC * HWDIM * HWDIM);   // n*d = 32
    dim3 grid(HWDIM / TW, HWDIM / TH, slices);               // (2, 8, 32)
    dim3 block(256);
    adaptive_conv_ps_kernel<<<grid, block, 0, stream>>>(burst, kernels, out);
}